// ClusterEncoder_35356170780707
// MI455X (gfx1250) — compile-verified
//
#include <hip/hip_runtime.h>

#define N_NODES 100000
#define N_EDGES 1600000
#define D_IN 64
#define D_H 256
#define MT 5            // 16-row M-subtiles per block: 100000 = 1250 * (16*MT)
#define BN_EPS 1e-5f

typedef __attribute__((ext_vector_type(2))) float v2f;
typedef __attribute__((ext_vector_type(8))) float v8f;

// Truncate a generic (flat) pointer to a __shared__ object to its 32-bit LDS
// byte offset (the LDS aperture keeps the offset in the low 32 bits).
__device__ __forceinline__ unsigned lds_addr32(const void* p) {
  return (unsigned)(unsigned long long)p;
}

// ---------------------------------------------------------------------------
// Stage 2: edge scatter-add.  aggr (=h0) was pre-initialized to x, so after
// this kernel h0[i] = x[i] + sum_{j in N(i)} x[j]  (GIN aggregate, eps=0).
// 16 threads per edge, each moves 4 floats (float4 gather + 4 f32 atomics).
// ---------------------------------------------------------------------------
__global__ void __launch_bounds__(256)
gin_scatter(const float* __restrict__ x, const int* __restrict__ src,
            const int* __restrict__ dst, float* __restrict__ aggr) {
  unsigned tid = blockIdx.x * 256u + threadIdx.x;
  unsigned e = tid >> 4;
  unsigned g = (tid & 15u) << 2;
  if (e >= N_EDGES) return;
  int s = src[e];
  int d = dst[e];
  const float4 v = *reinterpret_cast<const float4*>(x + (size_t)s * D_IN + g);
  float* p = aggr + (size_t)d * D_IN + g;
  atomicAdd(p + 0, v.x);
  atomicAdd(p + 1, v.y);
  atomicAdd(p + 2, v.z);
  atomicAdd(p + 3, v.w);
}

// ---------------------------------------------------------------------------
// Stages 3/4: out = relu(bn(A @ W + bias))  with A [N_NODES x K], W [K x 256].
// Block = 512 threads = 16 waves. Block stages MT*16 rows of A in LDS using
// GLOBAL_LOAD_ASYNC_TO_LDS_B128 (ASYNCcnt-tracked, no VGPR round-trip).
// Wave w owns output columns [16w, 16w+16); computes MT 16x16 tiles via
// V_WMMA_F32_16X16X4_F32, K-loop in steps of 4.
//
// f32 WMMA fragment layouts (ISA 7.12.2):
//   A 16x4 : lane<16 -> row=lane,    v0=A[r][k+0], v1=A[r][k+1]
//            lane>=16-> row=lane-16, v0=A[r][k+2], v1=A[r][k+3]
//   B 4x16 : lane<16 -> col=lane,    v0=B[k+0][c], v1=B[k+1][c]
//            lane>=16-> col=lane-16, v0=B[k+2][c], v1=B[k+3][c]
//   C/D    : vgpr r: lanes 0-15 -> (M=r, N=lane), lanes 16-31 -> (M=r+8, N=lane-16)
// ---------------------------------------------------------------------------
template <int K>
__global__ void __launch_bounds__(512)
gin_gemm_bn_relu(const float* __restrict__ A, const float* __restrict__ W,
                 const float* __restrict__ bias, const float* __restrict__ gamma,
                 const float* __restrict__ beta, const float* __restrict__ rmean,
                 const float* __restrict__ rvar, float* __restrict__ out) {
  __shared__ float sA[MT * 16 * K];

  const int tid = threadIdx.x;
  const int m0 = blockIdx.x * (MT * 16);

  // Rows m0..m0+79 are contiguous in row-major A: async-copy the whole slab
  // (MT*16*K floats) straight into LDS, 16 bytes per lane per issue.
  {
    const float* gbase = A + (size_t)m0 * K;             // uniform -> SGPR pair
    const unsigned sbase = lds_addr32((const void*)sA);  // LDS byte offset
    constexpr unsigned TOT16 = (unsigned)(MT * 16 * K / 4);  // # of 16B chunks
    for (unsigned i = tid; i < TOT16; i += 512) {
      unsigned lds_off = sbase + i * 16u;
      unsigned gv_off = i * 16u;
      asm volatile("global_load_async_to_lds_b128 %0, %1, %2"
                   :
                   : "v"(lds_off), "v"(gv_off), "s"(gbase)
                   : "memory");
    }
    asm volatile("s_wait_asynccnt 0x0" ::: "memory");
  }
  __syncthreads();

  const int wave = tid >> 5;
  const int lane = tid & 31;
  const int mrow = lane & 15;
  const int hi = lane >> 4;        // 0: lanes 0-15, 1: lanes 16-31
  const int koff = hi << 1;        // K sub-offset for A/B fragments
  const int col = (wave << 4) + mrow;

  v8f acc[MT];
  #pragma unroll
  for (int t = 0; t < MT; ++t) acc[t] = (v8f){0.f, 0.f, 0.f, 0.f, 0.f, 0.f, 0.f, 0.f};

  for (int k = 0; k < K; k += 4) {
    v2f b;
    b.x = W[(size_t)(k + koff) * D_H + col];
    b.y = W[(size_t)(k + koff + 1) * D_H + col];
    #pragma unroll
    for (int t = 0; t < MT; ++t) {
      const float* ap = &sA[(size_t)(t * 16 + mrow) * K + k + koff];
      v2f a;
      a.x = ap[0];
      a.y = ap[1];
      acc[t] = __builtin_amdgcn_wmma_f32_16x16x4_f32(
          /*neg_a=*/false, a, /*neg_b=*/false, b,
          /*c_mod=*/(short)0, acc[t], /*reuse_a=*/false, /*reuse_b=*/false);
    }
  }

  // BN + ReLU epilogue, folded per-column:
  //   bn(z) = (z - rm) * s + be, z = acc + bias  =>  acc*s + ((bias-rm)*s + be)
  const float s = gamma[col] * rsqrtf(rvar[col] + BN_EPS);
  const float t0 = (bias[col] - rmean[col]) * s + beta[col];

  #pragma unroll
  for (int t = 0; t < MT; ++t) {
    #pragma unroll
    for (int r = 0; r < 8; ++r) {
      const int row = m0 + t * 16 + r + (hi << 3);
      const float v = fmaxf(acc[t][r] * s + t0, 0.f);
      out[(size_t)row * D_H + col] = v;
    }
  }
}

extern "C" void kernel_launch(void* const* d_in, const int* in_sizes, int n_in,
                              void* d_out, int out_size, void* d_ws, size_t ws_size,
                              hipStream_t stream) {
  const float* x   = (const float*)d_in[0];
  // d_in[1] = pos (unused by reference)
  const int*   ei  = (const int*)d_in[2];      // [2, N_EDGES]: row0=src, row1=dst
  const float* W1  = (const float*)d_in[3];
  const float* b1  = (const float*)d_in[4];
  const float* g1  = (const float*)d_in[5];
  const float* be1 = (const float*)d_in[6];
  const float* rm1 = (const float*)d_in[7];
  const float* rv1 = (const float*)d_in[8];
  const float* W2  = (const float*)d_in[9];
  const float* b2  = (const float*)d_in[10];
  const float* g2  = (const float*)d_in[11];
  const float* be2 = (const float*)d_in[12];
  const float* rm2 = (const float*)d_in[13];
  const float* rv2 = (const float*)d_in[14];

  float* h0  = (float*)d_ws;                       // [N_NODES][64]  = 25.6 MB
  float* h1  = h0 + (size_t)N_NODES * D_IN;        // [N_NODES][256] = 102.4 MB
  float* out = (float*)d_out;

  // Stage 1: h0 = x (so scatter-add yields x + aggr directly).
  hipMemcpyAsync(h0, x, (size_t)N_NODES * D_IN * sizeof(float),
                 hipMemcpyDeviceToDevice, stream);

  // Stage 2: scatter-add over edges. 1.6M edges * 16 threads = 25.6M threads.
  gin_scatter<<<(N_EDGES * 16) / 256, 256, 0, stream>>>(x, ei, ei + N_EDGES, h0);

  // Stage 3: h1 = relu(bn1(h0 @ W1 + b1))
  gin_gemm_bn_relu<D_IN><<<N_NODES / (MT * 16), 512, 0, stream>>>(
      h0, W1, b1, g1, be1, rm1, rv1, h1);

  // Stage 4: out = relu(bn2(h1 @ W2 + b2))
  gin_gemm_bn_relu<D_H><<<N_NODES / (MT * 16), 512, 0, stream>>>(
      h1, W2, b2, g2, be2, rm2, rv2, out);
}